// GCN_81217831567578
// MI455X (gfx1250) — compile-verified
//
#include <hip/hip_runtime.h>
#include <math.h>

typedef __attribute__((ext_vector_type(2))) float v2f;
typedef __attribute__((ext_vector_type(8))) float v8f;

#define F_IN 512
#define HID  16
#define CLS  40

// ---------------------------------------------------------------------------
// init: deg = 1.0 (self-loop weight), agg1 = agg2 = 0
// ---------------------------------------------------------------------------
__global__ void k_init(float* __restrict__ deg, float* __restrict__ agg1,
                       float* __restrict__ agg2, int n) {
    int t = blockIdx.x * blockDim.x + threadIdx.x;
    if (t < n) deg[t] = 1.0f;
    if (t < n * HID) { agg1[t] = 0.0f; agg2[t] = 0.0f; }
}

// deg[col[e]] += w[e]
__global__ void k_degree(const int* __restrict__ col, const float* __restrict__ ew,
                         float* __restrict__ deg, int E) {
    int e = blockIdx.x * blockDim.x + threadIdx.x;
    if (e < E) atomicAdd(&deg[col[e]], ew[e]);
}

// deg -> dinv in place (deg >= 1 always due to self-loop)
__global__ void k_dinv(float* __restrict__ deg, int n) {
    int i = blockIdx.x * blockDim.x + threadIdx.x;
    if (i < n) {
        float d = deg[i];
        deg[i] = (d > 0.0f) ? (1.0f / sqrtf(d)) : 0.0f;
    }
}

// ---------------------------------------------------------------------------
// GEMM1: h1[N,16] = x[N,512] @ W1[512,16] via V_WMMA_F32_16X16X4_F32
// One wave per 16-row tile; K marched in steps of 4 (128 WMMAs per wave).
// A layout (16x4 f32): lane half h holds K=2h..2h+1 -> contiguous float2 load.
// D layout: VGPR r -> row m0+r (lanes 0-15) / m0+8+r (lanes 16-31), col=lane&15.
// ---------------------------------------------------------------------------
__global__ void __launch_bounds__(256) k_gemm1(const float* __restrict__ x,
                                               const float* __restrict__ W1,
                                               float* __restrict__ h1, int n) {
    __shared__ float lw[F_IN * HID];        // 32 KB of 320 KB LDS
    for (int i = threadIdx.x; i < F_IN * HID; i += 256) lw[i] = W1[i];
    __syncthreads();

    int wave   = (blockIdx.x * 256 + threadIdx.x) >> 5;
    int lane   = threadIdx.x & 31;
    int ntiles = (n + 15) >> 4;
    if (wave >= ntiles) return;             // uniform per-wave exit: EXEC stays full

    int m0   = wave << 4;
    int m    = lane & 15;                   // row-in-tile (A) / column (B, D)
    int half = lane >> 4;

    int arow_idx = m0 + m;
    if (arow_idx > n - 1) arow_idx = n - 1; // clamp (keeps EXEC all-ones)
    const float* arow = x + (size_t)arow_idx * F_IN + 2 * half;

    v8f c = {0.f, 0.f, 0.f, 0.f, 0.f, 0.f, 0.f, 0.f};

#pragma unroll 4
    for (int k = 0; k < F_IN; k += 4) {
        v2f a = __builtin_nontemporal_load((const v2f*)(arow + k)); // x streamed once
        int kk = k + 2 * half;
        v2f b;
        b.x = lw[kk * HID + m];
        b.y = lw[(kk + 1) * HID + m];
        c = __builtin_amdgcn_wmma_f32_16x16x4_f32(
                /*neg_a=*/false, a, /*neg_b=*/false, b,
                /*c_mod=*/(short)0, c, /*reuse_a=*/false, /*reuse_b=*/false);
    }

    // D store. Fast path (wave-uniform): full 16-row tile -> 8 plain b32 stores
    // at constant immediate offsets from one base address.
    float* orow = h1 + (size_t)(m0 + 8 * half) * HID + m;
    if (m0 + 16 <= n) {
#pragma unroll
        for (int r = 0; r < 8; ++r) orow[r * HID] = c[r];
    } else {
#pragma unroll
        for (int r = 0; r < 8; ++r) {
            int rr = m0 + r + 8 * half;
            if (rr < n) h1[(size_t)rr * HID + m] = c[r];
        }
    }
}

// ---------------------------------------------------------------------------
// Edge scatter: 16 lanes per edge (2 edges per wave), one feature each.
// agg[col,f] += h[row,f] * dinv[row]*w*dinv[col]   (L2-resident targets)
// ---------------------------------------------------------------------------
__global__ void k_agg(const int* __restrict__ row, const int* __restrict__ col,
                      const float* __restrict__ ew, const float* __restrict__ dinv,
                      const float* __restrict__ h, float* __restrict__ agg, int E) {
    long long t = (long long)blockIdx.x * blockDim.x + threadIdx.x;
    int e = (int)(t >> 4);
    if (e >= E) return;
    int f = (int)t & 15;
    int r = row[e];
    int c = col[e];
    float nrm = dinv[r] * ew[e] * dinv[c];
    atomicAdd(&agg[(size_t)c * HID + f], h[(size_t)r * HID + f] * nrm);
}

// z = relu(agg1 + h1 * dinv^2 (self-loop) + b1)
__global__ void k_combine1(const float* __restrict__ agg1, const float* __restrict__ h1,
                           const float* __restrict__ dinv, const float* __restrict__ b1,
                           float* __restrict__ z, int n) {
    int t = blockIdx.x * blockDim.x + threadIdx.x;
    if (t >= n * HID) return;
    int i = t >> 4, f = t & 15;
    float di = dinv[i];
    float v = agg1[t] + h1[t] * di * di + b1[f];
    z[t] = v > 0.0f ? v : 0.0f;
}

// ---------------------------------------------------------------------------
// Final: per node, a = agg2 + z*dinv^2; v = a @ W2 + b2; log_softmax(v).
// W2 (16x40) broadcast from LDS; 640 FMAs/node -> trivial vs memory.
// ---------------------------------------------------------------------------
__global__ void __launch_bounds__(256) k_final(const float* __restrict__ agg2,
                                               const float* __restrict__ z,
                                               const float* __restrict__ dinv,
                                               const float* __restrict__ W2,
                                               const float* __restrict__ b2,
                                               float* __restrict__ out, int n) {
    __shared__ float lw[HID * CLS];
    __shared__ float lb[CLS];
    for (int i = threadIdx.x; i < HID * CLS; i += 256) lw[i] = W2[i];
    if (threadIdx.x < CLS) lb[threadIdx.x] = b2[threadIdx.x];
    __syncthreads();

    int i = blockIdx.x * 256 + threadIdx.x;
    if (i >= n) return;

    float di = dinv[i];
    float d2 = di * di;

    float a[HID];
#pragma unroll
    for (int f = 0; f < HID; ++f)
        a[f] = agg2[(size_t)i * HID + f] + z[(size_t)i * HID + f] * d2;

    float v[CLS];
#pragma unroll
    for (int c = 0; c < CLS; ++c) v[c] = lb[c];
#pragma unroll
    for (int f = 0; f < HID; ++f) {
        float af = a[f];
#pragma unroll
        for (int c = 0; c < CLS; ++c) v[c] = fmaf(af, lw[f * CLS + c], v[c]);
    }

    float mx = v[0];
#pragma unroll
    for (int c = 1; c < CLS; ++c) mx = fmaxf(mx, v[c]);
    float s = 0.0f;
#pragma unroll
    for (int c = 0; c < CLS; ++c) s += expf(v[c] - mx);
    float ls = logf(s) + mx;
#pragma unroll
    for (int c = 0; c < CLS; ++c) out[(size_t)i * CLS + c] = v[c] - ls;
}

// ---------------------------------------------------------------------------
extern "C" void kernel_launch(void* const* d_in, const int* in_sizes, int n_in,
                              void* d_out, int out_size, void* d_ws, size_t ws_size,
                              hipStream_t stream) {
    const float* x  = (const float*)d_in[0];
    const int*   ei = (const int*)d_in[1];
    const float* ew = (const float*)d_in[2];
    const float* W1 = (const float*)d_in[3];
    const float* b1 = (const float*)d_in[4];
    const float* W2 = (const float*)d_in[5];
    const float* b2 = (const float*)d_in[6];
    float* out = (float*)d_out;

    int N = in_sizes[0] / F_IN;
    int E = in_sizes[2];
    const int* row = ei;      // edge_index[0]
    const int* col = ei + E;  // edge_index[1]

    // workspace layout (floats): deg[N] | h1[16N] | agg1[16N] | z[16N] | agg2[16N]
    float* ws   = (float*)d_ws;
    float* deg  = ws;                       // becomes dinv in place
    float* h1   = ws + (size_t)N;
    float* agg1 = ws + (size_t)17 * N;
    float* z    = ws + (size_t)33 * N;
    float* agg2 = ws + (size_t)49 * N;

    int nh = N * HID;

    k_init   <<<(nh + 255) / 256, 256, 0, stream>>>(deg, agg1, agg2, N);
    k_degree <<<(E + 255) / 256, 256, 0, stream>>>(col, ew, deg, E);
    k_dinv   <<<(N + 255) / 256, 256, 0, stream>>>(deg, N);

    int ntiles = (N + 15) / 16;
    k_gemm1  <<<(ntiles + 7) / 8, 256, 0, stream>>>(x, W1, h1, N);

    long long T = (long long)E * 16;
    unsigned aggBlocks = (unsigned)((T + 255) / 256);
    k_agg    <<<aggBlocks, 256, 0, stream>>>(row, col, ew, deg, h1, agg1, E);
    k_combine1<<<(nh + 255) / 256, 256, 0, stream>>>(agg1, h1, deg, b1, z, N);
    k_agg    <<<aggBlocks, 256, 0, stream>>>(row, col, ew, deg, z, agg2, E);
    k_final  <<<(N + 255) / 256, 256, 0, stream>>>(agg2, z, deg, W2, b2, out, N);
}